// ProductBasisBlock_73675868995698
// MI455X (gfx1250) — compile-verified
//
#include <hip/hip_runtime.h>

typedef float v2f __attribute__((ext_vector_type(2)));
typedef float v4f __attribute__((ext_vector_type(4)));
typedef float v8f __attribute__((ext_vector_type(8)));

#define CCH  256
#define IDIM 16
#define BDIM 4096
#define NK3  23
#define NK2  4
#define NK1  1

// workspace layout (float offsets)
#define WS_U3W 0                          // C * 4096   (M_c, interleaved for ds_load_b64)
#define WS_C2  (CCH * 4096)               // C * 256    (c2^T, interleaved)
#define WS_C1  (WS_C2 + CCH * 256)        // C * 16
#define WS_O3  (WS_C1 + CCH * 16)         // C * B      (out3 transposed [c][b])

// ---------------------------------------------------------------------------
// Stage A: fold per-channel weights into the CG bases.
//   M_c[(q*16+i), p] = sum_k U3[p,q,i,k] * w3[c,k]   stored interleaved:
//   flat[((kk>>1)*16 + p)*2 + (kk&1)]  so B-operand pairs (K=2t, K=2t+1)
//   for lanes of V_WMMA_F32_16X16X4_F32 are one 8-byte LDS load.
// ---------------------------------------------------------------------------
__launch_bounds__(256)
__global__ void mace_precompute(const float* __restrict__ U3,
                                const float* __restrict__ U2,
                                const float* __restrict__ U1,
                                const float* __restrict__ w3,
                                const float* __restrict__ w2,
                                const float* __restrict__ w1,
                                float* __restrict__ ws) {
    const int c = blockIdx.x;
    const int t = threadIdx.x;           // 256 threads
    const int p = t & 15;
    const int q = t >> 4;

    float w3r[NK3];
#pragma unroll
    for (int k = 0; k < NK3; ++k) w3r[k] = w3[c * NK3 + k];

    float* u3out = ws + WS_U3W + (size_t)c * 4096;
#pragma unroll
    for (int i = 0; i < 16; ++i) {
        float acc = 0.f;
        const float* u3p = U3 + ((p * 16 + q) * 16 + i) * NK3;
#pragma unroll
        for (int k = 0; k < NK3; ++k) acc += u3p[k] * w3r[k];
        const int kk = q * 16 + i;
        u3out[((kk >> 1) * 16 + p) * 2 + (kk & 1)] = acc;
    }

    { // c2^T[i,p] = sum_k U2[p,i,k] w2[c,k], interleaved over i
        const int i = t >> 4;
        float acc = 0.f;
#pragma unroll
        for (int k = 0; k < NK2; ++k) acc += U2[(p * 16 + i) * NK2 + k] * w2[c * NK2 + k];
        ws[WS_C2 + c * 256 + ((i >> 1) * 16 + p) * 2 + (i & 1)] = acc;
    }
    if (t < 16) {
        float acc = 0.f;
#pragma unroll
        for (int k = 0; k < NK1; ++k) acc += U1[t * NK1 + k] * w1[c * NK1 + k];
        ws[WS_C1 + c * 16 + t] = acc;
    }
}

// ---------------------------------------------------------------------------
// Stage B: symmetric contraction. grid = (C, 8), block = 256 (8 waves).
// Each wave handles 4 tiles of 16 nodes for one channel.
//   z~[b,p] = sum_{q,i} (x_q x_i) M_c[qi,p]   (64 WMMA K-steps)
//           + sum_i x_i c2T[i,p]              (4  WMMA K-steps)
//           + c1[p]
//   out3[b] = sum_p z~[b,p] x_p               (LDS transpose + shfl_xor 16)
// B-operands are tile-invariant: hoisted from LDS into VGPRs once per wave,
// so the hot loop is pure v_pk_mul -> v_wmma with no dscnt stalls.
// ---------------------------------------------------------------------------
__launch_bounds__(256, 1)
__global__ void mace_contract(const float* __restrict__ x,
                              const float* __restrict__ wsc,
                              float* __restrict__ o3) {
    __shared__ float sU3[4096];
    __shared__ float sC2[256];
    __shared__ float sC1[16];
    __shared__ float sZ[8 * 256];

    const int c = blockIdx.x;
    const int t = threadIdx.x;

    // stage M_c (16 KB), c2T (1 KB), c1 into LDS
    const float* gU3 = wsc + WS_U3W + (size_t)c * 4096;
#pragma unroll
    for (int j = 0; j < 4; ++j)
        ((v4f*)sU3)[t + j * 256] = ((const v4f*)gU3)[t + j * 256];
    sC2[t] = wsc[WS_C2 + c * 256 + t];
    if (t < 16) sC1[t] = wsc[WS_C1 + c * 16 + t];
    __syncthreads();

    const int w    = t >> 5;
    const int lane = t & 31;
    const int m    = lane & 15;     // A-layout: M = lane%16 for both halves
    const int hi   = lane >> 4;     // half selects K offset (+2)

    const float  c1v    = sC1[m];
    const float* u3lane = sU3 + m * 2 + hi * 32;   // + k*64 per K-step
    const float* c2lane = sC2 + m * 2 + hi * 32;
    float*       zrow   = sZ + w * 256;

    // hoist all B operands into registers (tile-invariant for this wave)
    v2f bU[64];
#pragma unroll
    for (int k = 0; k < 64; ++k) bU[k] = *(const v2f*)(u3lane + k * 64);
    v2f bC2[4];
#pragma unroll
    for (int k = 0; k < 4; ++k)  bC2[k] = *(const v2f*)(c2lane + k * 64);

    for (int it = 0; it < 4; ++it) {
        const int T  = blockIdx.y * 32 + w * 4 + it;  // b-tile 0..255
        const int b0 = T * 16;

        // full x row for b = b0 + m (both halves load the same row)
        const float* xrow = x + ((size_t)(b0 + m) * CCH + c) * IDIM;
        float xr[16];
#pragma unroll
        for (int j = 0; j < 4; ++j) {
            v4f v = ((const v4f*)xrow)[j];
            xr[j * 4 + 0] = v.x; xr[j * 4 + 1] = v.y;
            xr[j * 4 + 2] = v.z; xr[j * 4 + 3] = v.w;
        }
        // half-dependent K-slice selections: xs[2t+j] = x[4t + 2*hi + j]
        float xs[8];
#pragma unroll
        for (int tt = 0; tt < 4; ++tt) {
            xs[2 * tt]     = hi ? xr[4 * tt + 2] : xr[4 * tt];
            xs[2 * tt + 1] = hi ? xr[4 * tt + 3] : xr[4 * tt + 1];
        }
        // half-selected epilogue values: xh[j] = x[8*hi + j]  (compile-time
        // indices only -> keeps xr in VGPRs, no scratch spill)
        float xh[8];
#pragma unroll
        for (int j = 0; j < 8; ++j) xh[j] = hi ? xr[8 + j] : xr[j];

        v8f acc = {};
        // correlation-2 term: corr[b,p] = sum_i x_i * c2T[i,p]
#pragma unroll
        for (int k = 0; k < 4; ++k) {
            v2f a;  a.x = xs[2 * k]; a.y = xs[2 * k + 1];
            acc = __builtin_amdgcn_wmma_f32_16x16x4_f32(
                false, a, false, bC2[k], (short)0, acc, false, false);
        }
        // correlation-1 term: + c1[p]  (p = lane%16 across all rows)
#pragma unroll
        for (int r = 0; r < 8; ++r) acc[r] += c1v;

        // correlation-3 term: z[b,p] += sum_{kk}(x_q x_i) M_c[kk,p], K=256
#pragma unroll
        for (int k = 0; k < 64; ++k) {
            const int q  = k >> 2;        // compile-time per step
            const int tt = k & 3;
            v2f a;  a.x = xr[q] * xs[2 * tt];  a.y = xr[q] * xs[2 * tt + 1];
            acc = __builtin_amdgcn_wmma_f32_16x16x4_f32(
                false, a, false, bU[k], (short)0, acc, false, false);
        }

        // epilogue: out3[b] = sum_p z~[b,p] * x[b,p]
        // D layout -> row-per-lane via wave-private LDS transpose
#pragma unroll
        for (int r = 0; r < 8; ++r)
            zrow[(r + 8 * hi) * 16 + m] = acc[r];
        __asm__ volatile("s_wait_dscnt 0x0" ::: "memory");  // same-wave RAW
        float psum = 0.f;
        const float* zr = zrow + m * 16 + 8 * hi;           // lo: p=0..7, hi: p=8..15
#pragma unroll
        for (int j = 0; j < 8; ++j) psum += zr[j] * xh[j];
        psum += __shfl_xor(psum, 16, 32);                   // combine halves
        if (lane < 16) o3[(size_t)c * BDIM + b0 + m] = psum; // [c][b], coalesced
    }
}

// ---------------------------------------------------------------------------
// Stage C: out[b,d] = 0.0625 * sum_c out3[b,c] * W_lin[c,d]
// grid = (16 N-tiles, 32 M-groups), block = 256; W_lin columns staged in LDS.
// ---------------------------------------------------------------------------
__launch_bounds__(256)
__global__ void mace_linear(const float* __restrict__ o3,   // [C][B]
                            const float* __restrict__ Wl,   // [C][256]
                            float* __restrict__ out) {      // [B][256]
    __shared__ float sW[4096];
    const int t  = threadIdx.x;
    const int d0 = blockIdx.x * 16;
#pragma unroll
    for (int j4 = 0; j4 < 16; ++j4) {
        const int e   = t + j4 * 256;
        const int jj  = e & 1;
        const int n   = (e >> 1) & 15;
        const int K2i = e >> 5;
        sW[e] = Wl[(2 * K2i + jj) * 256 + d0 + n];
    }
    __syncthreads();

    const int w = t >> 5, lane = t & 31, m = lane & 15, hi = lane >> 4;
    const float* wlane = sW + m * 2 + hi * 32;
    const int b0 = (blockIdx.y * 8 + w) * 16;

    v8f acc = {};
#pragma unroll
    for (int k = 0; k < 64; ++k) {
        v2f a;
        a.x = o3[(size_t)(4 * k + 2 * hi)     * BDIM + b0 + m];
        a.y = o3[(size_t)(4 * k + 2 * hi + 1) * BDIM + b0 + m];
        v2f bb = *(const v2f*)(wlane + k * 64);
        acc = __builtin_amdgcn_wmma_f32_16x16x4_f32(
            false, a, false, bb, (short)0, acc, false, false);
    }
#pragma unroll
    for (int r = 0; r < 8; ++r)
        out[(size_t)(b0 + r + 8 * hi) * 256 + d0 + m] = acc[r] * 0.0625f;
}

// ---------------------------------------------------------------------------
extern "C" void kernel_launch(void* const* d_in, const int* in_sizes, int n_in,
                              void* d_out, int out_size, void* d_ws, size_t ws_size,
                              hipStream_t stream) {
    const float* x  = (const float*)d_in[0];  // (B,C,I)
    const float* U3 = (const float*)d_in[1];  // (I,I,I,K3)
    const float* U2 = (const float*)d_in[2];  // (I,I,K2)
    const float* U1 = (const float*)d_in[3];  // (I,K1)
    const float* w3 = (const float*)d_in[4];  // (C,K3)
    const float* w2 = (const float*)d_in[5];  // (C,K2)
    const float* w1 = (const float*)d_in[6];  // (C,K1)
    const float* Wl = (const float*)d_in[7];  // (C,C)
    float* out = (float*)d_out;               // (B,C)
    float* ws  = (float*)d_ws;

    mace_precompute<<<dim3(CCH), dim3(256), 0, stream>>>(U3, U2, U1, w3, w2, w1, ws);
    mace_contract  <<<dim3(CCH, 8), dim3(256), 0, stream>>>(x, ws, ws + WS_O3);
    mace_linear    <<<dim3(16, 32), dim3(256), 0, stream>>>(ws + WS_O3, Wl, out);
}